// SematicVoxelization_32057635897979
// MI455X (gfx1250) — compile-verified
//
#include <hip/hip_runtime.h>

// Problem constants (from reference)
#define B_      8
#define NV_     6890
#define NF_     13776
#define R_      128
#define R3_     (R_ * R_ * R_)         // 2,097,152
#define RADIUS_ 2
#define SIGMA_  0.01f
#define EPS0_   0.001f
#define VSIZE_  (2.0f / 128.0f)        // H_NORMALIZE / R
#define HALF_   63.5f                  // R/2 - 0.5
#define LOG2E_  1.4426950408889634f

typedef __attribute__((ext_vector_type(2))) float v2f;
typedef __attribute__((ext_vector_type(8))) float v8f;

// ---------------------------------------------------------------------------
// Zero-fill with B128 stores
// ---------------------------------------------------------------------------
__global__ __launch_bounds__(256) void zero4_kernel(float4* __restrict__ p, int n4) {
    int i = blockIdx.x * blockDim.x + threadIdx.x;
    int stride = gridDim.x * blockDim.x;
    float4 z; z.x = 0.f; z.y = 0.f; z.z = 0.f; z.w = 0.f;
    for (; i < n4; i += stride) p[i] = z;
}

// ---------------------------------------------------------------------------
// Face scatter: one wave32 per 16 faces of one batch.
// The full Gaussian exponent is computed on the matrix pipe:
//   D = A x B + C  with
//   A[m][k] = scaleL * (-2 off_x, -2 off_y, -2 off_z, |off|^2)   (16x4)
//   B[k][n] = (frac_x, frac_y, frac_z, 1)                        (4x16)
//   C[m][n] = scaleL * |frac|^2
// so D[m][n] = scaleL * |off - frac|^2 = log2-domain exponent, and the
// per-offset work collapses to one v_exp_f32 + 4 FP32 global atomics.
// Lanes 0-15 and 16-31 both gather face (lane&15), matching the WMMA
// B/C/D operand layouts (lo half: K=0,1 / M=0-7; hi half: K=2,3 / M=8-15).
// ---------------------------------------------------------------------------
__global__ __launch_bounds__(256) void scatter_kernel(
    const float* __restrict__ verts,   // (B, NV, 3)
    const float* __restrict__ vcode,   // (NV, 3)
    const int*   __restrict__ faces,   // (NF, 3)
    float*       __restrict__ sem,     // d_out: (B, 3, R^3) accumulators
    float*       __restrict__ wsum)    // d_ws : (B, R^3) accumulators
{
    const int lane = threadIdx.x & 31;
    const bool hi  = lane >= 16;
    const int fl16 = lane & 15;

    const int wave = blockIdx.x * (blockDim.x >> 5) + (threadIdx.x >> 5);
    const int gpb  = NF_ / 16;                 // 861 face-groups per batch
    const int b    = wave / gpb;
    const int g    = wave - b * gpb;
    const int f    = g * 16 + fl16;            // this lane's face

    // ---- gather face data (duplicated across wave halves; L2/WGP$ hits) ----
    const int i0 = faces[3 * f + 0];
    const int i1 = faces[3 * f + 1];
    const int i2 = faces[3 * f + 2];
    const float* vb = verts + (size_t)b * (NV_ * 3);
    const float third = 1.0f / 3.0f;

    const float cx = (vb[3 * i0 + 0] + vb[3 * i1 + 0] + vb[3 * i2 + 0]) * third;
    const float cy = (vb[3 * i0 + 1] + vb[3 * i1 + 1] + vb[3 * i2 + 1]) * third;
    const float cz = (vb[3 * i0 + 2] + vb[3 * i1 + 2] + vb[3 * i2 + 2]) * third;

    const float code0 = (vcode[3 * i0 + 0] + vcode[3 * i1 + 0] + vcode[3 * i2 + 0]) * third;
    const float code1 = (vcode[3 * i0 + 1] + vcode[3 * i1 + 1] + vcode[3 * i2 + 1]) * third;
    const float code2 = (vcode[3 * i0 + 2] + vcode[3 * i1 + 2] + vcode[3 * i2 + 2]) * third;

    // ---- voxel-space position, integer base, fractional part ----
    const float inv_vsize = 1.0f / VSIZE_;
    const float gx = cx * inv_vsize + HALF_;
    const float gy = cy * inv_vsize + HALF_;
    const float gz = cz * inv_vsize + HALF_;
    const float bxf = floorf(gx), byf = floorf(gy), bzf = floorf(gz);
    const int bx = (int)bxf, by = (int)byf, bz = (int)bzf;
    const float fx = gx - bxf, fy = gy - byf, fz = gz - bzf;
    const float frac2 = fx * fx + fy * fy + fz * fz;

    // log2-domain Gaussian scale: exp(-d2/(2 s^2)) = exp2(scaleL * |off-frac|^2)
    const float scaleL = -(VSIZE_ * VSIZE_) / (2.0f * SIGMA_ * SIGMA_) * LOG2E_;

    // B-operand (4x16): lo half supplies K=0,1 = (fx,fy); hi half K=2,3 = (fz,1)
    v2f bop;
    bop.x = hi ? fz : fx;
    bop.y = hi ? 1.0f : fy;

    // C-operand: broadcast scaleL*frac2 into all 8 accumulator slots
    const float cterm = scaleL * frac2;
    v8f cmat;
#pragma unroll
    for (int q = 0; q < 8; ++q) cmat[q] = cterm;

    float* semb = sem  + (size_t)b * 3 * R3_;
    float* wb   = wsum + (size_t)b * R3_;

    // 8 chunks of 16 offsets cover the 125-point (5x5x5) neighborhood
#pragma unroll
    for (int c = 0; c < 8; ++c) {
        // A-operand for M = fl16 in this chunk (same m for both wave halves)
        const int ma  = c * 16 + fl16;
        const int iax = ma / 25 - RADIUS_;
        const int iay = (ma / 5) % 5 - RADIUS_;
        const int iaz = ma % 5 - RADIUS_;
        const float off2a = (float)(iax * iax + iay * iay + iaz * iaz);
        v2f aop;
        aop.x = hi ? (scaleL * -2.0f * (float)iaz) : (scaleL * -2.0f * (float)iax);
        aop.y = hi ? (scaleL * off2a)              : (scaleL * -2.0f * (float)iay);

        // D[m][n] = scaleL * |off_m - frac_n|^2  (matrix pipe, co-executes)
        v8f dmat = __builtin_amdgcn_wmma_f32_16x16x4_f32(
            false, aop, false, bop, (short)0, cmat, false, false);

        // D layout: lane holds face (lane&15); VGPR v -> offset m = c*16+v (+8 hi)
#pragma unroll
        for (int v = 0; v < 8; ++v) {
            const int mlo = c * 16 + v;        // compile-time constant
            const int mhi = mlo + 8;           // compile-time constant
            const bool lo_ok = (mlo < 125);    // always true (mlo <= 119)
            const bool hi_ok = (mhi < 125);    // false only for c=7, v>=5
            // offset components: select between two compile-time constants
            const int oi = hi ? (mhi / 25 - RADIUS_)      : (mlo / 25 - RADIUS_);
            const int oj = hi ? ((mhi / 5) % 5 - RADIUS_) : ((mlo / 5) % 5 - RADIUS_);
            const int ok = hi ? (mhi % 5 - RADIUS_)       : (mlo % 5 - RADIUS_);
            const bool mval = hi ? hi_ok : lo_ok;

            const int vx = bx + oi, vy = by + oj, vz = bz + ok;
            if (mval &
                ((unsigned)vx < (unsigned)R_) &
                ((unsigned)vy < (unsigned)R_) &
                ((unsigned)vz < (unsigned)R_)) {
                const float w = __builtin_amdgcn_exp2f(dmat[v]);   // v_exp_f32
                const int xyz = (vx << 14) + (vy << 7) + vz;
                // hardware FP32 global atomics (no CAS loops)
                unsafeAtomicAdd(wb + xyz, w);
                unsafeAtomicAdd(semb + xyz,           w * code0);
                unsafeAtomicAdd(semb + R3_ + xyz,     w * code1);
                unsafeAtomicAdd(semb + 2 * R3_ + xyz, w * code2);
            }
        }
    }
}

// ---------------------------------------------------------------------------
// Normalize: out[b,c,xyz] = sem[b,c,xyz] / (wsum[b,xyz] + EPS0), float4-wide
// ---------------------------------------------------------------------------
__global__ __launch_bounds__(256) void normalize_kernel(
    float* __restrict__ out, const float* __restrict__ wsum)
{
    const int n4 = B_ * R3_ / 4;
    int i = blockIdx.x * blockDim.x + threadIdx.x;
    if (i >= n4) return;
    const int per_b = R3_ / 4;                  // 524288 float4 per batch
    const int b  = i / per_b;
    const int x4 = i - b * per_b;

    const float4 wv = ((const float4*)wsum)[(size_t)b * per_b + x4];
    float4 inv;
    inv.x = 1.0f / (wv.x + EPS0_);
    inv.y = 1.0f / (wv.y + EPS0_);
    inv.z = 1.0f / (wv.z + EPS0_);
    inv.w = 1.0f / (wv.w + EPS0_);

    float4* ob = (float4*)out + (size_t)b * 3 * per_b + x4;
#pragma unroll
    for (int ch = 0; ch < 3; ++ch) {
        float4 s = ob[(size_t)ch * per_b];
        s.x *= inv.x; s.y *= inv.y; s.z *= inv.z; s.w *= inv.w;
        ob[(size_t)ch * per_b] = s;
    }
}

// ---------------------------------------------------------------------------
extern "C" void kernel_launch(void* const* d_in, const int* in_sizes, int n_in,
                              void* d_out, int out_size, void* d_ws, size_t ws_size,
                              hipStream_t stream) {
    const float* verts = (const float*)d_in[0];   // (B, NV, 3)
    const float* vcode = (const float*)d_in[1];   // (NV, 3)
    const int*   faces = (const int*)d_in[2];     // (NF, 3)
    // d_in[3] (occ_volume) is all-zeros and unused by the math.

    float* out = (float*)d_out;                   // (B, 3, R, R, R)
    float* ws  = (float*)d_ws;                    // (B, R^3) wsum accumulators

    const int n_out4 = B_ * 3 * R3_ / 4;          // 12,582,912 float4
    const int n_ws4  = B_ * R3_ / 4;              //  4,194,304 float4
    zero4_kernel<<<4096, 256, 0, stream>>>((float4*)out, n_out4);
    zero4_kernel<<<2048, 256, 0, stream>>>((float4*)ws, n_ws4);

    const int waves  = B_ * (NF_ / 16);           // 6888 waves, all full
    const int blocks = waves / 8;                 // 861 blocks x 8 wave32
    scatter_kernel<<<blocks, 256, 0, stream>>>(verts, vcode, faces, out, ws);

    const int n4 = B_ * R3_ / 4;
    normalize_kernel<<<(n4 + 255) / 256, 256, 0, stream>>>(out, ws);
}